// SAGE_11484742549903
// MI455X (gfx1250) — compile-verified
//
#include <hip/hip_runtime.h>
#include <hip/hip_bf16.h>

#define NN 100000
#define EE 600000
#define DD 128
#define LDA 132   // padded LDS row stride (floats): 132%64=4 -> conflict-free, float4-aligned

typedef float v2f __attribute__((ext_vector_type(2)));
typedef float v8f __attribute__((ext_vector_type(8)));

// ---------------- utility kernels ----------------

__global__ __launch_bounds__(256) void sage_zero_f(float* __restrict__ p, int n) {
    int i = blockIdx.x * 256 + threadIdx.x;
    if (i < n) p[i] = 0.0f;
}

__global__ __launch_bounds__(256) void sage_zero_f4(float4* __restrict__ p, int n4) {
    int i = blockIdx.x * 256 + threadIdx.x;
    if (i < n4) p[i] = make_float4(0.f, 0.f, 0.f, 0.f);
}

__global__ __launch_bounds__(256) void sage_deg(const int* __restrict__ dst,
                                                float* __restrict__ deg) {
    int e = blockIdx.x * 256 + threadIdx.x;
    if (e < EE) atomicAdd(&deg[dst[e]], 1.0f);
}

__global__ __launch_bounds__(256) void sage_invdeg(float* __restrict__ deg) {
    int i = blockIdx.x * 256 + threadIdx.x;
    if (i < NN) {
        float d = deg[i];
        deg[i] = (d > 0.0f) ? (1.0f / d) : 0.0f;
    }
}

// ---------------- aggregation: agg[dst] += h[src] ----------------
// one 128-thread block per edge: scalar src/dst loads, coalesced 512B row gather,
// atomic f32 scatter-add (both tables are L2-resident on MI455X: 51.2MB << 192MB)
__global__ __launch_bounds__(128) void sage_agg(const float* __restrict__ h,
                                                const int* __restrict__ src,
                                                const int* __restrict__ dst,
                                                float* __restrict__ agg) {
    int e = blockIdx.x;
    int c = threadIdx.x;
    int s = src[e];
    int d = dst[e];
    float v = h[(size_t)s * DD + c];
    atomicAdd(&agg[(size_t)d * DD + c], v);
}

// ---------------- fused dual-GEMM + bias + relu ----------------
// Out[16 rows x 128 cols per block] = relu?( h@Ws + (agg*inv_deg)@Wn + b )
// 8 wave32 per block; each wave owns one 16x16 output tile; 32 fp32-WMMA steps
// per weight matrix (K=128, 4 per step).
__global__ __launch_bounds__(256) void sage_gemm(const float* __restrict__ A_self,
                                                 const float* __restrict__ A_agg,
                                                 const float* __restrict__ inv_deg,
                                                 const float* __restrict__ Ws,
                                                 const float* __restrict__ Wn,
                                                 const float* __restrict__ bias,
                                                 float* __restrict__ Out,
                                                 int relu) {
    __shared__ float sH[16 * LDA];
    __shared__ float sG[16 * LDA];

    const int tid  = threadIdx.x;
    const int row0 = blockIdx.x * 16;

    // cooperative stage of the 16x128 h-tile and agg-tile into LDS (float4)
    // 512 float4 per matrix, 256 threads -> 2 each
    for (int r = 0; r < 2; ++r) {
        int i   = tid + r * 256;        // 0..511
        int row = i >> 5;               // /32 (32 float4 per row)
        int c4  = (i & 31) << 2;        // *4
        float4 hv = *(const float4*)(A_self + (size_t)(row0 + row) * DD + c4);
        float4 gv = *(const float4*)(A_agg  + (size_t)(row0 + row) * DD + c4);
        *(float4*)&sH[row * LDA + c4] = hv;
        *(float4*)&sG[row * LDA + c4] = gv;
    }
    __syncthreads();

    const int wave   = tid >> 5;        // 0..7 -> 16-col tile
    const int lane   = tid & 31;
    const int row_in = lane & 15;       // A-matrix M row per lane
    const int khalf  = lane >> 4;       // 0: K=0,1  1: K=2,3 (per 4-K step)
    const int col    = lane & 15;       // B/C column within tile
    const int colg   = wave * 16 + col; // global column 0..127

    const float inv = inv_deg[row0 + row_in];

    v8f acc = {};
    #pragma unroll
    for (int k0 = 0; k0 < DD; k0 += 4) {
        int ka = k0 + khalf * 2;
        // A fragment (16x4 f32): lane row_in holds K=ka, ka+1
        v2f ah; ah.x = sH[row_in * LDA + ka];
                ah.y = sH[row_in * LDA + ka + 1];
        // B fragment (4x16 f32): lane col holds rows K=ka, ka+1
        v2f bw; bw.x = Ws[(size_t)ka * DD + colg];
                bw.y = Ws[(size_t)(ka + 1) * DD + colg];
        acc = __builtin_amdgcn_wmma_f32_16x16x4_f32(false, ah, false, bw,
                                                    (short)0, acc, false, false);
        v2f ag; ag.x = sG[row_in * LDA + ka] * inv;
                ag.y = sG[row_in * LDA + ka + 1] * inv;
        v2f bn; bn.x = Wn[(size_t)ka * DD + colg];
                bn.y = Wn[(size_t)(ka + 1) * DD + colg];
        acc = __builtin_amdgcn_wmma_f32_16x16x4_f32(false, ag, false, bn,
                                                    (short)0, acc, false, false);
    }

    // epilogue: C/D layout -> VGPR v holds M = v (lanes 0-15) / v+8 (lanes 16-31)
    const float bc = bias[colg];
    const int rbase = row0 + khalf * 8;
    #pragma unroll
    for (int v = 0; v < 8; ++v) {
        float o = acc[v] + bc;
        if (relu) o = fmaxf(o, 0.0f);
        Out[(size_t)(rbase + v) * DD + colg] = o;
    }
}

// ---------------- driver ----------------

extern "C" void kernel_launch(void* const* d_in, const int* in_sizes, int n_in,
                              void* d_out, int out_size, void* d_ws, size_t ws_size,
                              hipStream_t stream) {
    const float* feat = (const float*)d_in[0];
    const int*   src  = (const int*)d_in[1];
    const int*   dst  = (const int*)d_in[2];
    const float* Ws[3] = {(const float*)d_in[3], (const float*)d_in[6], (const float*)d_in[9]};
    const float* Wn[3] = {(const float*)d_in[4], (const float*)d_in[7], (const float*)d_in[10]};
    const float* bb[3] = {(const float*)d_in[5], (const float*)d_in[8], (const float*)d_in[11]};

    // workspace carve-up (256B aligned)
    char* ws = (char*)d_ws;
    size_t off = 0;
    float* inv = (float*)(ws + off); off += ((size_t)NN * 4 + 255) & ~(size_t)255;
    float* agg = (float*)(ws + off); off += (size_t)NN * DD * 4;
    float* h1  = (float*)(ws + off); off += (size_t)NN * DD * 4;
    float* h2  = (float*)(ws + off);
    float* out = (float*)d_out;

    const int ND  = NN * DD;           // 12,800,000
    const int ND4 = ND / 4;

    // in-degree -> 1/deg (re-zeroed every call: atomics accumulate)
    sage_zero_f<<<(NN + 255) / 256, 256, 0, stream>>>(inv, NN);
    sage_deg   <<<(EE + 255) / 256, 256, 0, stream>>>(dst, inv);
    sage_invdeg<<<(NN + 255) / 256, 256, 0, stream>>>(inv);

    const float* hin = feat;
    float* houts[3] = {h1, h2, out};
    for (int l = 0; l < 3; ++l) {
        sage_zero_f4<<<(ND4 + 255) / 256, 256, 0, stream>>>((float4*)agg, ND4);
        sage_agg<<<EE, 128, 0, stream>>>(hin, src, dst, agg);
        sage_gemm<<<NN / 16, 256, 0, stream>>>(hin, agg, inv, Ws[l], Wn[l], bb[l],
                                               houts[l], (l < 2) ? 1 : 0);
        hin = houts[l];
    }
}